// PFNet7_71648644432478
// MI455X (gfx1250) — compile-verified
//
#include <hip/hip_runtime.h>
#include <hip/hip_bf16.h>

#define N_PTS 16384

typedef __attribute__((ext_vector_type(2))) float v2f;
typedef __attribute__((ext_vector_type(8))) float v8f;

__device__ __forceinline__ float elu_f(float v)   { return v > 0.0f ? v : expm1f(v); }
__device__ __forceinline__ float lrelu_f(float v) { return v >= 0.0f ? v : 0.01f * v; }

// ---------------------------------------------------------------------------
// Kernel 1: nn1 MLP  12 -> 64 -> 64 -> 12 with ELU, thread per point.
// Weights staged in LDS; accumulators fully register-resident (const indices).
// ---------------------------------------------------------------------------
__global__ __launch_bounds__(256)
void nn1_kernel(const float* __restrict__ x,
                const float* __restrict__ W1, const float* __restrict__ b1,
                const float* __restrict__ W2, const float* __restrict__ b2,
                const float* __restrict__ W3, const float* __restrict__ b3,
                float* __restrict__ h12)
{
    __shared__ float sW1[12 * 64], sB1[64], sW2[64 * 64], sB2[64], sW3[64 * 12], sB3[12];
    for (int t = threadIdx.x; t < 12 * 64; t += 256) sW1[t] = W1[t];
    for (int t = threadIdx.x; t < 64;      t += 256) { sB1[t] = b1[t]; sB2[t] = b2[t]; }
    for (int t = threadIdx.x; t < 64 * 64; t += 256) sW2[t] = W2[t];
    for (int t = threadIdx.x; t < 64 * 12; t += 256) sW3[t] = W3[t];
    if (threadIdx.x < 12) sB3[threadIdx.x] = b3[threadIdx.x];
    __syncthreads();

    const int i = blockIdx.x * 256 + threadIdx.x;

    float xi[12];
#pragma unroll
    for (int d = 0; d < 12; ++d) xi[d] = x[i * 12 + d];

    // layer1 (ELU) fused into layer2 accumulation
    float a2[64];
#pragma unroll
    for (int o = 0; o < 64; ++o) a2[o] = sB2[o];
    for (int j = 0; j < 64; ++j) {
        float h = sB1[j];
#pragma unroll
        for (int d = 0; d < 12; ++d) h += xi[d] * sW1[d * 64 + j];
        h = elu_f(h);
#pragma unroll
        for (int o = 0; o < 64; ++o) a2[o] += h * sW2[j * 64 + o];
    }

    // layer2 (ELU) fused into layer3 accumulation
    float a3[12];
#pragma unroll
    for (int o = 0; o < 12; ++o) a3[o] = sB3[o];
#pragma unroll
    for (int j = 0; j < 64; ++j) {
        float h = elu_f(a2[j]);
#pragma unroll
        for (int o = 0; o < 12; ++o) a3[o] += h * sW3[j * 12 + o];
    }
#pragma unroll
    for (int o = 0; o < 12; ++o) h12[i * 12 + o] = a3[o];
}

// ---------------------------------------------------------------------------
// Kernel 2: s = h@Ws+bs  [N,4],  hf = h@Wh+bh  [N,22],  sn = |s|^2
// ---------------------------------------------------------------------------
__global__ __launch_bounds__(256)
void space_feat_kernel(const float* __restrict__ h12,
                       const float* __restrict__ Ws, const float* __restrict__ bs,
                       const float* __restrict__ Wh, const float* __restrict__ bh,
                       float* __restrict__ s4, float* __restrict__ sn, float* __restrict__ hf)
{
    const int i = blockIdx.x * 256 + threadIdx.x;
    float hi[12];
#pragma unroll
    for (int d = 0; d < 12; ++d) hi[d] = h12[i * 12 + d];

    float q = 0.0f;
#pragma unroll
    for (int o = 0; o < 4; ++o) {
        float a = bs[o];
#pragma unroll
        for (int d = 0; d < 12; ++d) a += hi[d] * Ws[d * 4 + o];
        s4[i * 4 + o] = a;
        q += a * a;
    }
    sn[i] = q;

#pragma unroll
    for (int o = 0; o < 22; ++o) {
        float a = bh[o];
#pragma unroll
        for (int d = 0; d < 12; ++d) a += hi[d] * Wh[d * 22 + o];
        hf[i * 22 + o] = a;
    }
}

// ---------------------------------------------------------------------------
// Kernel 3: brute-force kNN (K=16) in 4-dim learned space.
// d2 = sn_i + sn_j - 2*dot (same formula as reference, same selection set).
// Per-thread register-resident sorted top-16, LDS-chunked candidates.
// ---------------------------------------------------------------------------
__global__ __launch_bounds__(256)
void knn_kernel(const float* __restrict__ s4, const float* __restrict__ sn,
                int* __restrict__ idx_out, float* __restrict__ d2_out)
{
    __shared__ float cs[256][4];
    __shared__ float csn[256];

    const int i = blockIdx.x * 256 + threadIdx.x;
    const float si0 = s4[i * 4 + 0], si1 = s4[i * 4 + 1];
    const float si2 = s4[i * 4 + 2], si3 = s4[i * 4 + 3];
    const float sni = sn[i];

    float bd[16]; int bi[16];
#pragma unroll
    for (int k = 0; k < 16; ++k) { bd[k] = 3.4e38f; bi[k] = 0; }

    for (int c0 = 0; c0 < N_PTS; c0 += 256) {
        __syncthreads();
        {
            const int j = c0 + threadIdx.x;
            cs[threadIdx.x][0] = s4[j * 4 + 0];
            cs[threadIdx.x][1] = s4[j * 4 + 1];
            cs[threadIdx.x][2] = s4[j * 4 + 2];
            cs[threadIdx.x][3] = s4[j * 4 + 3];
            csn[threadIdx.x]   = sn[j];
        }
        __syncthreads();

        for (int t = 0; t < 256; ++t) {
            const float dot = si0 * cs[t][0] + si1 * cs[t][1] + si2 * cs[t][2] + si3 * cs[t][3];
            const float d2  = sni + csn[t] - 2.0f * dot;
            if (d2 < bd[15]) {                 // strictly better than current worst
                float cd = d2; int ci = c0 + t;
#pragma unroll
                for (int q = 0; q < 16; ++q) { // bubble into sorted position
                    if (cd <= bd[q]) {
                        float td = bd[q]; int ti = bi[q];
                        bd[q] = cd; bi[q] = ci; cd = td; ci = ti;
                    }
                }
            }
        }
    }
#pragma unroll
    for (int k = 0; k < 16; ++k) {
        idx_out[i * 16 + k] = bi[k];
        d2_out[i * 16 + k]  = bd[k];
    }
}

// ---------------------------------------------------------------------------
// Kernel 4: message aggregation (mean || max of w*h_n), lin_out1 + lin_out2,
// leaky_relu -> h64
// ---------------------------------------------------------------------------
__global__ __launch_bounds__(256)
void agg_kernel(const float* __restrict__ h12, const float* __restrict__ hf,
                const int* __restrict__ knn_i, const float* __restrict__ knn_d,
                const float* __restrict__ W1,                       // 12 x 64 (no bias)
                const float* __restrict__ W2, const float* __restrict__ b2, // 44 x 64
                float* __restrict__ h64)
{
    __shared__ float sW1[12 * 64], sW2[44 * 64], sB2[64];
    for (int t = threadIdx.x; t < 12 * 64; t += 256) sW1[t] = W1[t];
    for (int t = threadIdx.x; t < 44 * 64; t += 256) sW2[t] = W2[t];
    for (int t = threadIdx.x; t < 64;      t += 256) sB2[t] = b2[t];
    __syncthreads();

    const int i = blockIdx.x * 256 + threadIdx.x;

    float sum[22], mx[22];
#pragma unroll
    for (int c = 0; c < 22; ++c) { sum[c] = 0.0f; mx[c] = -3.4e38f; }

    for (int k = 0; k < 16; ++k) {
        const int   j = knn_i[i * 16 + k];
        const float w = expf(-10.0f * knn_d[i * 16 + k]);
#pragma unroll
        for (int c = 0; c < 22; ++c) {
            const float m = hf[j * 22 + c] * w;
            sum[c] += m;
            mx[c]  = fmaxf(mx[c], m);
        }
    }
#pragma unroll
    for (int c = 0; c < 22; ++c) sum[c] *= (1.0f / 16.0f); // mean

    float xi[12];
#pragma unroll
    for (int d = 0; d < 12; ++d) xi[d] = h12[i * 12 + d];

#pragma unroll
    for (int o = 0; o < 64; ++o) {
        float a = sB2[o];
#pragma unroll
        for (int d = 0; d < 12; ++d) a += xi[d] * sW1[d * 64 + o];
#pragma unroll
        for (int c = 0; c < 22; ++c) a += sum[c] * sW2[c * 64 + o];
#pragma unroll
        for (int c = 0; c < 22; ++c) a += mx[c] * sW2[(22 + c) * 64 + o];
        h64[i * 64 + o] = lrelu_f(a);
    }
}

// ---------------------------------------------------------------------------
// Kernel 5: build nn3 input: [h64 (64) | pred_ids (6) | x0 (12) | pad (2)] = 84
// ---------------------------------------------------------------------------
__global__ __launch_bounds__(256)
void concat_kernel(const float* __restrict__ h64, const float* __restrict__ ids,
                   const float* __restrict__ x, float* __restrict__ xc)
{
    const int i = blockIdx.x * 256 + threadIdx.x;
    float* r = xc + (size_t)i * 84;
#pragma unroll
    for (int o = 0; o < 64; ++o) r[o] = h64[i * 64 + o];
#pragma unroll
    for (int o = 0; o < 6; ++o) r[64 + o] = ids[i * 6 + o];
#pragma unroll
    for (int o = 0; o < 12; ++o) r[70 + o] = x[i * 12 + o];
    r[82] = 0.0f; r[83] = 0.0f;
}

// ---------------------------------------------------------------------------
// Kernel 6: generic fused 4-layer MLP  kin -> 256 -> 256 -> 256 -> 6
// (ELU between hidden layers) using V_WMMA_F32_16X16X4_F32.
// Block = 64 threads = 2 waves; each wave owns a 16-row tile (32 rows/block).
// Activations ping-pong between two 32x256 LDS buffers (64 KB total).
// fp32 A 16x4 layout: lanes 0-15 -> M=lane, K base 0; lanes 16-31 -> K base 2;
// frag.x = K(base), frag.y = K(base+1).  B 4x16: rows striped across lane
// halves the same way, col = lane&15.  C/D: VGPR j -> M = j + 8*(lane>=16),
// N = lane&15.
// ---------------------------------------------------------------------------
__global__ __launch_bounds__(64)
void mlp4_wmma_kernel(const float* __restrict__ Xin, int kin,
                      const float* __restrict__ W1, const float* __restrict__ B1,
                      const float* __restrict__ W2, const float* __restrict__ B2,
                      const float* __restrict__ W3, const float* __restrict__ B3,
                      const float* __restrict__ W4, const float* __restrict__ B4,
                      float* __restrict__ out)     // [N, 6]
{
    __shared__ float buf[2][32][256];               // 64 KB ping-pong
    const int tid    = threadIdx.x;
    const int wave   = tid >> 5;
    const int lane   = tid & 31;
    const int laneLo = lane & 15;
    const int laneHi = lane >> 4;                   // 0 or 1
    const int wrow   = wave * 16;
    const int rowBlk = blockIdx.x * 32;

    // stage the input tile (32 x kin)
    for (int t = tid; t < 32 * kin; t += 64) {
        const int r = t / kin, c = t - r * kin;
        buf[0][r][c] = Xin[(size_t)(rowBlk + r) * kin + c];
    }
    __syncthreads();

    // hidden layer: src -> dst, kout = 256, W row-stride 256
    auto layer = [&](int src, int dst, const float* __restrict__ W,
                     const float* __restrict__ B, int ksteps) {
        for (int ct = 0; ct < 16; ++ct) {
            v8f acc;
            const float bias = B[ct * 16 + laneLo];
#pragma unroll
            for (int j = 0; j < 8; ++j) acc[j] = bias;
            for (int k = 0; k < ksteps; ++k) {
                const int kk = k * 4 + laneHi * 2;
                v2f a, b;
                a.x = buf[src][wrow + laneLo][kk];
                a.y = buf[src][wrow + laneLo][kk + 1];
                const int col = ct * 16 + laneLo;
                b.x = W[(size_t)kk * 256 + col];
                b.y = W[(size_t)(kk + 1) * 256 + col];
                acc = __builtin_amdgcn_wmma_f32_16x16x4_f32(
                        false, a, false, b, (short)0, acc, false, false);
            }
#pragma unroll
            for (int j = 0; j < 8; ++j)
                buf[dst][wrow + j + laneHi * 8][ct * 16 + laneLo] = elu_f(acc[j]);
        }
        __syncthreads();
    };

    layer(0, 1, W1, B1, kin / 4);   // kin -> 256
    layer(1, 0, W2, B2, 64);        // 256 -> 256
    layer(0, 1, W3, B3, 64);        // 256 -> 256

    // final layer: 256 -> 6 (single 16-wide col tile, cols 0..5 valid)
    {
        v8f acc;
        const float bias = (laneLo < 6) ? B4[laneLo] : 0.0f;
#pragma unroll
        for (int j = 0; j < 8; ++j) acc[j] = bias;
        for (int k = 0; k < 64; ++k) {
            const int kk = k * 4 + laneHi * 2;
            v2f a, b;
            a.x = buf[1][wrow + laneLo][kk];
            a.y = buf[1][wrow + laneLo][kk + 1];
            b.x = (laneLo < 6) ? W4[kk * 6 + laneLo] : 0.0f;
            b.y = (laneLo < 6) ? W4[(kk + 1) * 6 + laneLo] : 0.0f;
            acc = __builtin_amdgcn_wmma_f32_16x16x4_f32(
                    false, a, false, b, (short)0, acc, false, false);
        }
        if (laneLo < 6) {
#pragma unroll
            for (int j = 0; j < 8; ++j) {
                const int r = rowBlk + wrow + j + laneHi * 8;
                out[(size_t)r * 6 + laneLo] = acc[j];
            }
        }
    }
}

// ---------------------------------------------------------------------------
// Launch. Input order (jax pytree leaves; None bias of lin_out1 is dropped):
//  0:x  1:nn1W1 2:nn1b1 3:nn1W2 4:nn1b2 5:nn1W3 6:nn1b3
//  7:Ws 8:bs  9:Wh 10:bh  11:lin_out1 W  12:lin_out2 W 13:lin_out2 b
//  14..21: nn2 (W,b)x4   22..29: nn3 (W,b)x4
// ---------------------------------------------------------------------------
extern "C" void kernel_launch(void* const* d_in, const int* in_sizes, int n_in,
                              void* d_out, int out_size, void* d_ws, size_t ws_size,
                              hipStream_t stream)
{
    (void)in_sizes; (void)n_in; (void)out_size; (void)ws_size;
    const int N = N_PTS;
    const float* F[30];
    for (int i = 0; i < 30; ++i) F[i] = (const float*)d_in[i];

    float* ws = (float*)d_ws;
    size_t off = 0;
    float* h12 = ws + off; off += (size_t)N * 12;
    float* s4  = ws + off; off += (size_t)N * 4;
    float* snv = ws + off; off += (size_t)N;
    float* hf  = ws + off; off += (size_t)N * 22;
    float* kd2 = ws + off; off += (size_t)N * 16;
    int*   kix = (int*)(ws + off); off += (size_t)N * 16;
    float* h64 = ws + off; off += (size_t)N * 64;
    float* xc  = ws + off; off += (size_t)N * 84;

    float* pred_ids = (float*)d_out;                 // [N,6]
    float* pred_p4  = (float*)d_out + (size_t)N * 6; // [N,6]

    dim3 b256(256), g64(N / 256);

    nn1_kernel<<<g64, b256, 0, stream>>>(F[0], F[1], F[2], F[3], F[4], F[5], F[6], h12);
    space_feat_kernel<<<g64, b256, 0, stream>>>(h12, F[7], F[8], F[9], F[10], s4, snv, hf);
    knn_kernel<<<g64, b256, 0, stream>>>(s4, snv, kix, kd2);
    agg_kernel<<<g64, b256, 0, stream>>>(h12, hf, kix, kd2, F[11], F[12], F[13], h64);
    mlp4_wmma_kernel<<<N / 32, 64, 0, stream>>>(h64, 64,
        F[14], F[15], F[16], F[17], F[18], F[19], F[20], F[21], pred_ids);
    concat_kernel<<<g64, b256, 0, stream>>>(h64, pred_ids, F[0], xc);
    mlp4_wmma_kernel<<<N / 32, 64, 0, stream>>>(xc, 84,
        F[22], F[23], F[24], F[25], F[26], F[27], F[28], F[29], pred_p4);
}